// GcnDiag_7919919694205
// MI455X (gfx1250) — compile-verified
//
#include <hip/hip_runtime.h>

typedef float v2f __attribute__((ext_vector_type(2)));
typedef float v8f __attribute__((ext_vector_type(8)));

// ---------------- degree / normalization precompute ----------------

__global__ void k_init_deg(float* __restrict__ deg, int n) {
  int i = blockIdx.x * blockDim.x + threadIdx.x;
  if (i < n) deg[i] = 1.0f;  // self-loop fill value
}

__global__ void k_deg_accum(const int* __restrict__ dst, const float* __restrict__ w,
                            float* __restrict__ deg, int E) {
  int e = blockIdx.x * blockDim.x + threadIdx.x;
  if (e < E) unsafeAtomicAdd(&deg[dst[e]], w[e]);
}

__global__ void k_dinv(float* __restrict__ deg_dinv, float* __restrict__ self_norm, int n) {
  int i = blockIdx.x * blockDim.x + threadIdx.x;
  if (i < n) {
    float d  = deg_dinv[i];
    float di = (d > 0.0f) ? rsqrtf(d) : 0.0f;
    deg_dinv[i]  = di;
    self_norm[i] = di * di;
  }
}

__global__ void k_norm(const int* __restrict__ src, const int* __restrict__ dst,
                       const float* __restrict__ w, const float* __restrict__ dinv,
                       float* __restrict__ norm, int E) {
  int e = blockIdx.x * blockDim.x + threadIdx.x;
  if (e < E) norm[e] = dinv[src[e]] * w[e] * dinv[dst[e]];
}

// ---------------- dense GEMM: H = X @ W via v_wmma_f32_16x16x4_f32 ----------------
// blockDim = (32,4): 4 wave32, each owns a 16-row tile; all waves share one
// 16-column tile of W, staged zero-padded into LDS once per block.
// LDS layout interleaves K-pairs: element (k, n) lives at (k>>1)*32 + n*2 + (k&1),
// so a lane's (ka, ka+1) operand pair is one aligned ds_load_b64.
//
// Branch-free body:
//  - A rows clamped to N-1 (out-of-range rows feed only unstored D rows)
//  - K-tail handled by mask-multiply (v_cndmask), never predicated loads
//  - EXEC stays all-1s throughout, as WMMA requires.
#define KPAD_MAX 80  // max in-dim (80), already a multiple of 4

__global__ void k_gemm_wmma(const float* __restrict__ X, const float* __restrict__ W,
                            float* __restrict__ H, int N, int K, int OUT) {
  __shared__ float Wlds[KPAD_MAX * 16];

  const int lane    = threadIdx.x;                                  // 0..31
  const int rowBase = (blockIdx.x * blockDim.y + threadIdx.y) * 16; // wave's M tile
  const int colBase = blockIdx.y * 16;                              // block's N tile
  const int m     = lane & 15;
  const int khalf = (lane >> 4) * 2;  // lanes 0-15: K=0,1 ; lanes 16-31: K=2,3
  const int col = colBase + m;

  // ---- stage W tile into LDS (zero-padded in both K and col) ----
  const int Kpad = (K + 3) & ~3;
  const int tid  = threadIdx.y * 32 + lane;
  for (int idx = tid; idx < Kpad * 16; idx += 128) {
    const int k  = idx >> 4;
    const int nn = idx & 15;
    float v = 0.0f;
    if (k < K && (colBase + nn) < OUT) v = W[k * OUT + colBase + nn];
    Wlds[((k >> 1) << 5) + (nn << 1) + (k & 1)] = v;
  }
  __syncthreads();

  const int row   = rowBase + m;
  const int rowc  = (row < N) ? row : (N - 1);   // clamp: garbage rows never stored
  const int aBase = rowc * K;                    // 32-bit: N*K <= 10M
  const int bLane = (m << 1);                    // lane's column slot in LDS layout
  const int Kfloor = K & ~3;

  v8f acc = {};
  for (int k0 = 0; k0 < Kfloor; k0 += 4) {       // unpredicated main loop
    const int ka = k0 + khalf;
    v2f a, b;
    a.x = X[aBase + ka];
    a.y = X[aBase + ka + 1];
    b   = *(const v2f*)&Wlds[((ka >> 1) << 5) + bLane];
    acc = __builtin_amdgcn_wmma_f32_16x16x4_f32(false, a, false, b,
                                                (short)0, acc, false, false);
  }
  if (Kfloor < K) {  // K tail (layer 1: K=1) — clamped loads, mask-multiply
    const int ka  = Kfloor + khalf;
    const int ia  = (ka     < K) ? ka     : (K - 1);
    const int ib  = (ka + 1 < K) ? ka + 1 : (K - 1);
    const float fa = (ka     < K) ? 1.0f : 0.0f;
    const float fb = (ka + 1 < K) ? 1.0f : 0.0f;
    v2f a, b;
    a.x = X[aBase + ia] * fa;
    a.y = X[aBase + ib] * fb;
    b   = *(const v2f*)&Wlds[((ka >> 1) << 5) + bLane];
    acc = __builtin_amdgcn_wmma_f32_16x16x4_f32(false, a, false, b,
                                                (short)0, acc, false, false);
  }

  // C/D layout: VGPR r holds M = r (lanes 0-15) / M = r+8 (lanes 16-31), N = lane&15
  const int  mrow     = rowBase + (lane >> 4) * 8;
  const bool fullTile = ((rowBase + 16) <= N) & ((colBase + 16) <= OUT);
  if (fullTile) {
#pragma unroll
    for (int r = 0; r < 8; ++r) H[(mrow + r) * OUT + col] = acc[r];
  } else {
    const bool colOk = col < OUT;
#pragma unroll
    for (int r = 0; r < 8; ++r) {
      const int rr = mrow + r;
      if (rr < N && colOk) H[rr * OUT + col] = acc[r];
    }
  }
}

// ---------------- misc elementwise ----------------

__global__ void k_zero(float* __restrict__ p, long long n) {
  long long i = (long long)blockIdx.x * blockDim.x + threadIdx.x;
  if (i < n) p[i] = 0.0f;
}

// agg[dst[e], c] += h[src[e], c] * norm[e] — one thread per (edge, col); consecutive
// threads hit consecutive columns of one row (coalesced), atomics resolve in L2.
__global__ void k_scatter(const float* __restrict__ H, const int* __restrict__ src,
                          const int* __restrict__ dst, const float* __restrict__ norm,
                          float* __restrict__ AGG, int E, int OUT) {
  long long tid = (long long)blockIdx.x * blockDim.x + threadIdx.x;
  long long total = (long long)E * OUT;
  if (tid >= total) return;
  int e = (int)(tid / OUT);
  int c = (int)(tid - (long long)e * OUT);
  float v = H[src[e] * OUT + c] * norm[e];          // 32-bit: N*OUT <= 10M
  unsafeAtomicAdd(&AGG[dst[e] * OUT + c], v);
}

// out = agg + h * self_norm[row] + bias[col]  (+ ReLU on last layer)
__global__ void k_combine(const float* __restrict__ H, const float* __restrict__ self_norm,
                          const float* __restrict__ AGG, const float* __restrict__ bias,
                          float* __restrict__ out, int N, int OUT, int relu) {
  long long tid = (long long)blockIdx.x * blockDim.x + threadIdx.x;
  long long total = (long long)N * OUT;
  if (tid >= total) return;
  int i = (int)tid;
  int r = i / OUT;
  int c = i - r * OUT;
  float v = AGG[i] + H[i] * self_norm[r] + bias[c];
  if (relu) v = fmaxf(v, 0.0f);
  out[i] = v;
}

// ---------------- host-side orchestration ----------------

extern "C" void kernel_launch(void* const* d_in, const int* in_sizes, int n_in,
                              void* d_out, int out_size, void* d_ws, size_t ws_size,
                              hipStream_t stream) {
  (void)n_in; (void)out_size; (void)ws_size;

  const float* x  = (const float*)d_in[0];
  const int*   ei = (const int*)d_in[1];
  const float* ew = (const float*)d_in[2];
  const int N = in_sizes[0];      // x is [N,1]
  const int E = in_sizes[2];      // edge_weights is [E]
  const int* src = ei;            // edge_index[0]
  const int* dst = ei + E;        // edge_index[1]

  // workspace layout (floats)
  float* ws        = (float*)d_ws;
  float* dinv      = ws;                         // N (deg, then deg^-1/2)
  float* self_norm = dinv + N;                   // N
  float* norm      = self_norm + N;              // E
  float* buf[3];
  buf[0] = norm + E;                             // N*100 each
  buf[1] = buf[0] + (size_t)N * 100;
  buf[2] = buf[1] + (size_t)N * 100;

  const int dims[6] = {1, 20, 40, 60, 80, 100};

  const int TB = 256;
  k_init_deg <<<(N + TB - 1) / TB, TB, 0, stream>>>(dinv, N);
  k_deg_accum<<<(E + TB - 1) / TB, TB, 0, stream>>>(dst, ew, dinv, E);
  k_dinv     <<<(N + TB - 1) / TB, TB, 0, stream>>>(dinv, self_norm, N);
  k_norm     <<<(E + TB - 1) / TB, TB, 0, stream>>>(src, dst, ew, dinv, norm, E);

  const float* cur = x;
  int which = 0;
  for (int L = 0; L < 5; ++L) {
    const int K   = dims[L];
    const int OUT = dims[L + 1];
    const float* W = (const float*)d_in[3 + 2 * L];
    const float* b = (const float*)d_in[4 + 2 * L];

    float* h   = buf[which];
    float* agg = buf[(which + 1) % 3];

    dim3 gblk(32, 4);                                  // 4 wave32 per block
    dim3 ggrid((N + 63) / 64, (OUT + 15) / 16);
    k_gemm_wmma<<<ggrid, gblk, 0, stream>>>(cur, W, h, N, K, OUT);

    long long tot  = (long long)N * OUT;
    long long etot = (long long)E * OUT;
    k_zero   <<<(unsigned)((tot  + TB - 1) / TB), TB, 0, stream>>>(agg, tot);
    k_scatter<<<(unsigned)((etot + TB - 1) / TB), TB, 0, stream>>>(h, src, dst, norm, agg, E, OUT);

    float* dest = (L == 4) ? (float*)d_out : agg;
    k_combine<<<(unsigned)((tot + TB - 1) / TB), TB, 0, stream>>>(h, self_norm, agg, b,
                                                                  dest, N, OUT, L == 4 ? 1 : 0);
    cur   = agg;                 // combine wrote agg in place (or d_out on last layer)
    which = (which + 2) % 3;     // next h reuses old buffer, keeps cur intact
  }
}